// OP_Attention_69277822485310
// MI455X (gfx1250) — compile-verified
//
#include <hip/hip_runtime.h>
#include <math.h>

#define B_ 8
#define C_ 64
#define N_ 256
#define F_ 128
#define RS_ 132          // padded LDS row stride (floats): 4*mrow banks -> conflict-free WMMA frags
#define EPS_ 1e-5f

typedef float v2f __attribute__((ext_vector_type(2)));
typedef float v8f __attribute__((ext_vector_type(8)));

// ---------------------------------------------------------------------------
// gfx1250 async global->LDS copy (ASYNCcnt-tracked, bypasses VGPRs).
// Inline asm per bridge doc (portable across ROCm7.2 / amdgpu-toolchain).
// ---------------------------------------------------------------------------
__device__ __forceinline__ void async_load_f4(unsigned lds_byte_off, const float* gaddr) {
    asm volatile("global_load_async_to_lds_b128 %0, %1, off"
                 :: "v"(lds_byte_off), "v"(gaddr)
                 : "memory");
}

__device__ __forceinline__ void wait_async0() {
#if __has_builtin(__builtin_amdgcn_s_wait_asynccnt)
    __builtin_amdgcn_s_wait_asynccnt(0);
#else
    asm volatile("s_wait_asynccnt 0x0" ::: "memory");
#endif
}

// ---------------------------------------------------------------------------
// Block-wide sum reduction (256 threads). Returns result to all threads.
// ---------------------------------------------------------------------------
__device__ __forceinline__ float block_reduce(float v, float* s_red) {
    const int tid = threadIdx.x;
    s_red[tid] = v;
    __syncthreads();
#pragma unroll
    for (int off = 128; off > 0; off >>= 1) {
        if (tid < off) s_red[tid] += s_red[tid + off];
        __syncthreads();
    }
    float r = s_red[0];
    __syncthreads();
    return r;
}

// ---------------------------------------------------------------------------
// Kernel 1: one block per (b,c). Async-stage the 256x128 f32 tile into LDS,
// compute gating mask via V_WMMA_F32_16X16X4_F32, masked group sums, softmax
// weights, and weighted BN partial sums. h is read from HBM exactly once.
// ---------------------------------------------------------------------------
__global__ __launch_bounds__(256) void k1_stats(
    const float* __restrict__ h,   const float* __restrict__ op,
    const float* __restrict__ opS, const float* __restrict__ opQ,
    float* __restrict__ wsW, unsigned* __restrict__ wsMask,
    float* __restrict__ wsS1, float* __restrict__ wsS2)
{
    __shared__ float    s_tile[N_ * RS_];   // 256 rows x 132 floats (pad 4) = 132 KB
    __shared__ float    s_opB[F_];          // op[c, Fd:2Fd]
    __shared__ unsigned s_mb[N_ / 32];      // 256 mask bits
    __shared__ float    s_red[256];
    __shared__ float    s_colS[256];
    __shared__ float    s_colQ[256];

    const int tid = threadIdx.x;
    const int bc  = blockIdx.x;          // b*C + c
    const int c   = bc & (C_ - 1);
    const float* __restrict__ hb   = h   + (size_t)bc * (N_ * F_);
    const float* __restrict__ opc  = op  + c * (2 * F_);
    const float* __restrict__ opSc = opS + c * (2 * F_);
    const float* __restrict__ opQc = opQ + c * (2 * F_);

    // ---- Async-stage h[b,c,:,:] into LDS (coalesced float4, padded rows) ---
    const unsigned tile_base = (unsigned)(size_t)(const void*)&s_tile[0];
    for (int i = tid; i < (N_ * F_) / 4; i += 256) {      // 8192 float4s, 32/thread
        const int row = i >> 5;          // token n
        const int col = i & 31;          // float4 within feature dim
        async_load_f4(tile_base + (unsigned)(row * RS_ + col * 4) * 4u,
                      hb + (size_t)i * 4);
    }
    if (tid < F_)      s_opB[tid] = opc[F_ + tid];
    if (tid < N_ / 32) s_mb[tid]  = 0u;
    wait_async0();          // each wave drains its own ASYNCcnt
    __syncthreads();        // publish LDS tile to all waves

    // s0 = source . op[c, 0:Fd]   (source row == tile row 0)
    float p0 = (tid < F_) ? s_tile[tid] * opc[tid] : 0.f;
    const float s0 = block_reduce(p0, s_red);

    // ---- Pass A: 256 gating dots via V_WMMA_F32_16X16X4_F32 ----------------
    // A 16x4 f32: lane<16 -> K{0,1}, lane>=16 -> K{2,3}, M = lane%16 (ISA 7.12.2).
    // B = op chunk replicated over all 16 columns -> every D column is the dot.
    // acc seeded with s0 so D = s0 + h[row].opB directly.
    const int wave = tid >> 5;
    const int lane = tid & 31;
    const int hh   = lane >> 4;
    const int mrow = lane & 15;

    for (int t = wave; t < 16; t += 8) {
        const float* rowp = &s_tile[(size_t)(t * 16 + mrow) * RS_];
        v8f acc;
#pragma unroll
        for (int i = 0; i < 8; ++i) acc[i] = s0;
#pragma unroll 4
        for (int k0 = 0; k0 < F_; k0 += 4) {
            const float2 av = *(const float2*)(rowp  + k0 + 2 * hh);   // ds_load
            const float2 bv = *(const float2*)(s_opB + k0 + 2 * hh);   // ds_load
            v2f a; a[0] = av.x; a[1] = av.y;
            v2f b; b[0] = bv.x; b[1] = bv.y;
            acc = __builtin_amdgcn_wmma_f32_16x16x4_f32(
                false, a, false, b, (short)0, acc, false, false);
        }
        if (mrow == 0) {   // lanes 0/16 own rows v / v+8 of the tile (C/D layout)
            unsigned m8 = 0u;
#pragma unroll
            for (int v = 0; v < 8; ++v) m8 |= (acc[v] >= 0.f) ? (1u << v) : 0u;
            atomicOr(&s_mb[t >> 1], m8 << ((t & 1) * 16 + hh * 8));
        }
    }
    __syncthreads();

    // ---- Pass B: masked column/group sums straight from LDS ----------------
    const int f  = tid & (F_ - 1);
    const int nh = tid >> 7;
    float cS = 0.f, cQ = 0.f, cS2 = 0.f, cQ2 = 0.f;
    for (int n = nh; n < N_; n += 2) {
        float v = s_tile[(size_t)n * RS_ + f];
        bool  m = (s_mb[n >> 5] >> (n & 31)) & 1u;
        float vs = m ? v : 0.f;
        float vq = m ? 0.f : v;
        cS += vs; cQ += vq; cS2 += vs * vs; cQ2 += vq * vq;
    }
    s_colS[tid] = cS;
    s_colQ[tid] = cQ;

    const float totS  = block_reduce(cS,  s_red);   // barriers publish s_col*
    const float totQ  = block_reduce(cQ,  s_red);
    const float totS2 = block_reduce(cS2, s_red);
    const float totQ2 = block_reduce(cQ2, s_red);

    // supp_attn / query_attn dots (source row is tile row 0)
    float psa = 0.f, pqa = 0.f;
    if (tid < F_) {
        float sm = (s_colS[tid] + s_colS[tid + F_]) * (1.f / N_);
        float qm = (s_colQ[tid] + s_colQ[tid + F_]) * (1.f / N_);
        psa = sm * opSc[tid] + s_tile[tid] * opSc[F_ + tid];
        pqa = qm * opQc[tid] + s_tile[tid] * opQc[F_ + tid];
    }
    const float sa = block_reduce(psa, s_red);
    const float qa = block_reduce(pqa, s_red);

    if (tid == 0) {
        // 2-way softmax == sigmoid of score difference (stable)
        float w_s = 1.f / (1.f + expf(qa - sa));
        float w_q = 1.f / (1.f + expf(sa - qa));
        wsW[bc * 2 + 0] = w_s;
        wsW[bc * 2 + 1] = w_q;
        wsS1[bc] = w_s * totS + w_q * totQ;                      // sum(h_opn)
        wsS2[bc] = w_s * w_s * totS2 + w_q * w_q * totQ2;        // sum(h_opn^2)
    }
    if (tid < N_ / 32) wsMask[bc * (N_ / 32) + tid] = s_mb[tid];
}

// ---------------------------------------------------------------------------
// Kernel 2: per-channel BN stats -> fused scale/shift.
// ---------------------------------------------------------------------------
__global__ void k2_norm(const float* __restrict__ wsS1, const float* __restrict__ wsS2,
                        const float* __restrict__ gamma, const float* __restrict__ beta,
                        float* __restrict__ wsScale, float* __restrict__ wsShift)
{
    const int c = threadIdx.x;   // 64 threads
    float S1 = 0.f, S2 = 0.f;
#pragma unroll
    for (int b = 0; b < B_; ++b) {
        S1 += wsS1[b * C_ + c];
        S2 += wsS2[b * C_ + c];
    }
    const float invM = 1.f / ((float)B_ * (float)N_ * (float)F_);
    float mean = S1 * invM;
    float var  = S2 * invM - mean * mean;
    float sc   = rsqrtf(var + EPS_) * gamma[c];
    wsScale[c] = sc;
    wsShift[c] = beta[c] - mean * sc;
}

// ---------------------------------------------------------------------------
// Kernel 3: out = elu(h * w(mask) * scale[c] + shift[c]), float4 streams.
// h re-read is L2-resident (64 MB < 192 MB L2); 64 MB streamed store.
// ---------------------------------------------------------------------------
__global__ __launch_bounds__(256) void k3_apply(
    const float* __restrict__ h, const float* __restrict__ wsW,
    const unsigned* __restrict__ wsMask, const float* __restrict__ wsScale,
    const float* __restrict__ wsShift, float* __restrict__ out)
{
    __shared__ float    sw[4];
    __shared__ unsigned smb[N_ / 32];
    const int bc  = blockIdx.x >> 5;
    const int sub = blockIdx.x & 31;
    const int c   = bc & (C_ - 1);
    if (threadIdx.x == 0) {
        sw[0] = wsW[bc * 2 + 0];
        sw[1] = wsW[bc * 2 + 1];
        sw[2] = wsScale[c];
        sw[3] = wsShift[c];
    }
    if (threadIdx.x < N_ / 32) smb[threadIdx.x] = wsMask[bc * (N_ / 32) + threadIdx.x];
    __syncthreads();

    const int lin = sub * 256 + threadIdx.x;   // float4 index within (b,c)
    const int n   = lin >> 5;                  // token
    const int fq  = lin & 31;                  // float4 within feature dim
    const size_t base = ((size_t)bc * N_ + n) * F_ + (size_t)fq * 4;

    float4 v = *(const float4*)(h + base);
    const bool  m     = (smb[n >> 5] >> (n & 31)) & 1u;
    const float scale = (m ? sw[0] : sw[1]) * sw[2];
    const float shift = sw[3];

    float4 o; float y;
    y = v.x * scale + shift; o.x = (y > 0.f) ? y : expm1f(y);
    y = v.y * scale + shift; o.y = (y > 0.f) ? y : expm1f(y);
    y = v.z * scale + shift; o.z = (y > 0.f) ? y : expm1f(y);
    y = v.w * scale + shift; o.w = (y > 0.f) ? y : expm1f(y);
    *(float4*)(out + base) = o;
}

// ---------------------------------------------------------------------------
extern "C" void kernel_launch(void* const* d_in, const int* in_sizes, int n_in,
                              void* d_out, int out_size, void* d_ws, size_t ws_size,
                              hipStream_t stream)
{
    const float* h     = (const float*)d_in[0];
    const float* op    = (const float*)d_in[1];
    const float* opS   = (const float*)d_in[2];
    const float* opQ   = (const float*)d_in[3];
    const float* gamma = (const float*)d_in[4];
    const float* beta  = (const float*)d_in[5];
    float* out = (float*)d_out;

    // workspace carve-up (~25 KB total)
    float*    wsW     = (float*)d_ws;                           // 2*B*C
    unsigned* wsMask  = (unsigned*)(wsW + 2 * B_ * C_);         // B*C*(N/32)
    float*    wsS1    = (float*)(wsMask + B_ * C_ * (N_ / 32)); // B*C
    float*    wsS2    = wsS1 + B_ * C_;                         // B*C
    float*    wsScale = wsS2 + B_ * C_;                         // C
    float*    wsShift = wsScale + C_;                           // C

    k1_stats<<<B_ * C_, 256, 0, stream>>>(h, op, opS, opQ, wsW, wsMask, wsS1, wsS2);
    k2_norm<<<1, C_, 0, stream>>>(wsS1, wsS2, gamma, beta, wsScale, wsShift);
    k3_apply<<<B_ * C_ * 32, 256, 0, stream>>>(h, wsW, wsMask, wsScale, wsShift, out);

    (void)in_sizes; (void)n_in; (void)out_size; (void)ws_size;
}